// SegBrightnessLoss_84731114816126
// MI455X (gfx1250) — compile-verified
//
#include <hip/hip_runtime.h>
#include <hip/hip_bf16.h>

#define NCLS 11
#define NPIX (16u * 512u * 512u)   // 4,194,304 pixels
#define PLANE (512u * 512u)        // 262,144 (2^18)
#define S1_BLOCKS 1024
#define S1_THREADS 256

typedef float v8f __attribute__((ext_vector_type(8)));
typedef float v2f __attribute__((ext_vector_type(2)));
typedef float f4  __attribute__((ext_vector_type(4)));
typedef int   i4  __attribute__((ext_vector_type(4)));

// ---------------------------------------------------------------------------
// Stage 1: streaming per-class reduction at the HBM roofline.
// Each thread handles 4 sweeps x 4 consecutive pixels (float4/int4 loads).
// Per pixel: xm = mean over 3 channels; accumulate S[c], C[c] (class c = y,
// remapped to dead class 15 when xm == 0) and Q = sum(xm^2).
// Block-level reduction is a deterministic serial sum through LDS.
// ---------------------------------------------------------------------------
__global__ __launch_bounds__(S1_THREADS)
void seg_stage1(const float* __restrict__ x, const int* __restrict__ y,
                float* __restrict__ partS, float* __restrict__ partC,
                float* __restrict__ partQ)
{
    float s[NCLS], n[NCLS], q = 0.0f;
#pragma unroll
    for (int c = 0; c < NCLS; ++c) { s[c] = 0.0f; n[c] = 0.0f; }

    const unsigned tid = blockIdx.x * S1_THREADS + threadIdx.x;

#pragma unroll
    for (int sweep = 0; sweep < 4; ++sweep) {
        const unsigned p  = tid * 4u + ((unsigned)sweep << 20);  // grid covers 2^20 px/sweep
        const unsigned b  = p >> 18;                              // batch index
        const unsigned hw = p & (PLANE - 1u);
        const float* xb = x + (size_t)(b * 3u) * PLANE + hw;
        f4 x0 = *(const f4*)(xb);
        f4 x1 = *(const f4*)(xb + PLANE);
        f4 x2 = *(const f4*)(xb + 2u * PLANE);
        i4 yv = *(const i4*)(y + p);
#pragma unroll
        for (int r = 0; r < 4; ++r) {
            const float xm = (x0[r] + x1[r] + x2[r]) * (1.0f / 3.0f);
            q = fmaf(xm, xm, q);
            const int cls2 = (xm != 0.0f) ? yv[r] : 15;  // dead class 15 for zeros
#pragma unroll
            for (int c = 0; c < NCLS; ++c) {
                const float m = (cls2 == c) ? 1.0f : 0.0f;
                s[c] = fmaf(m, xm, s[c]);
                n[c] += m;
            }
        }
    }

    // Deterministic block reduction through LDS (serial sums, fixed order).
    __shared__ float red[2 * NCLS + 1][S1_THREADS];
#pragma unroll
    for (int c = 0; c < NCLS; ++c) {
        red[c][threadIdx.x]        = s[c];
        red[NCLS + c][threadIdx.x] = n[c];
    }
    red[2 * NCLS][threadIdx.x] = q;
    __syncthreads();

    const int t = threadIdx.x;
    if (t < 16) {
        float acc = 0.0f;
        if (t < NCLS)
            for (int i = 0; i < S1_THREADS; ++i) acc += red[t][i];
        partS[blockIdx.x * 16 + t] = acc;          // classes 11..15 padded with 0
    } else if (t < 32) {
        const int c = t - 16;
        float acc = 0.0f;
        if (c < NCLS)
            for (int i = 0; i < S1_THREADS; ++i) acc += red[NCLS + c][i];
        partC[blockIdx.x * 16 + c] = acc;
    } else if (t == 32) {
        float acc = 0.0f;
        for (int i = 0; i < S1_THREADS; ++i) acc += red[2 * NCLS][i];
        partQ[blockIdx.x] = acc;
    }
}

// ---------------------------------------------------------------------------
// Stage 2: reduce 1024 block-partials with V_WMMA_F32_16X16X4_F32.
// A = partial rows (row m = class = lane%16, K slots = block index),
// B = all-ones => D[m][n] = sum_k A[m][k] = per-class total, replicated in n.
// All-ones B makes the K slot->K mapping irrelevant (sum over all slots).
// 8 waves x 128 blocks each (2 chained WMMAs per 4 blocks), then a
// deterministic serial cross-wave combine and a float64 final formula.
// ---------------------------------------------------------------------------
__global__ __launch_bounds__(256)
void seg_stage2(const float* __restrict__ partS, const float* __restrict__ partC,
                const float* __restrict__ partQ, float* __restrict__ out)
{
    const int lane = threadIdx.x & 31;
    const int wave = threadIdx.x >> 5;
    const int lm   = lane & 15;
    const int hi   = lane >> 4;   // half-wave selects K block {0,1} vs {2,3}

    v8f accS = {0.f, 0.f, 0.f, 0.f, 0.f, 0.f, 0.f, 0.f};
    v8f accC = {0.f, 0.f, 0.f, 0.f, 0.f, 0.f, 0.f, 0.f};
    v2f ones; ones[0] = 1.0f; ones[1] = 1.0f;

#pragma unroll 4
    for (int g = 0; g < 32; ++g) {
        const int blk = wave * 128 + g * 4 + hi * 2;   // K slot -> block id
        v2f a, b;
        a[0] = partS[(blk + 0) * 16 + lm];
        a[1] = partS[(blk + 1) * 16 + lm];
        b[0] = partC[(blk + 0) * 16 + lm];
        b[1] = partC[(blk + 1) * 16 + lm];
        accS = __builtin_amdgcn_wmma_f32_16x16x4_f32(false, a, false, ones,
                                                     (short)0, accS, false, false);
        accC = __builtin_amdgcn_wmma_f32_16x16x4_f32(false, b, false, ones,
                                                     (short)0, accC, false, false);
    }

    // D layout: VGPR r, lanes 0-15 -> row r (col = lane), lanes 16-31 -> row r+8.
    // All columns are identical (B == ones); take column 0 of each half.
    __shared__ float lS[8][16];
    __shared__ float lC[8][16];
    __shared__ double lQ[32];
    if (lm == 0) {
#pragma unroll
        for (int r = 0; r < 8; ++r) {
            lS[wave][r + 8 * hi] = accS[r];
            lC[wave][r + 8 * hi] = accC[r];
        }
    }
    __syncthreads();

    if (threadIdx.x < 32) {
        double acc = 0.0;
        for (int i = (int)threadIdx.x; i < S1_BLOCKS; i += 32) acc += (double)partQ[i];
        lQ[threadIdx.x] = acc;
    }
    __syncthreads();

    if (threadIdx.x == 0) {
        double Qt = 0.0;
        for (int i = 0; i < 32; ++i) Qt += lQ[i];
        const double N = (double)NPIX;
        double sum2 = 0.0, sum2c = 0.0;
        for (int c = 0; c < NCLS; ++c) {
            double S = 0.0, C = 0.0;
            for (int w = 0; w < 8; ++w) {
                S += (double)lS[w][c];
                C += (double)lC[w][c];
            }
            sum2  += S * S;
            sum2c += S * S * C;
        }
        const double d = (Qt - (2.0 / N) * sum2 + (1.0 / (N * N)) * sum2c) / N;
        out[0] = (float)d;
    }
}

extern "C" void kernel_launch(void* const* d_in, const int* in_sizes, int n_in,
                              void* d_out, int out_size, void* d_ws, size_t ws_size,
                              hipStream_t stream)
{
    (void)in_sizes; (void)n_in; (void)out_size; (void)ws_size;
    const float* x = (const float*)d_in[0];
    const int*   y = (const int*)d_in[1];

    float* partS = (float*)d_ws;              // 1024 * 16 floats
    float* partC = partS + S1_BLOCKS * 16;    // 1024 * 16 floats
    float* partQ = partC + S1_BLOCKS * 16;    // 1024 floats   (~132 KB total)

    seg_stage1<<<S1_BLOCKS, S1_THREADS, 0, stream>>>(x, y, partS, partC, partQ);
    seg_stage2<<<1, 256, 0, stream>>>(partS, partC, partQ, (float*)d_out);
}